// Chebnet_81398220194150
// MI455X (gfx1250) — compile-verified
//
#include <hip/hip_runtime.h>

typedef float v2f __attribute__((ext_vector_type(2)));
typedef float v8f __attribute__((ext_vector_type(8)));

#define N_NODES 50000
#define N_EDGES 600000
#define DIM     128
#define NF      (N_NODES * DIM)   // 6,400,000 floats per feature matrix

// ---------------------------------------------------------------------------
// Utility: fill a float buffer with a constant
// ---------------------------------------------------------------------------
__global__ void k_fill(float* __restrict__ p, float v, int n) {
    int i = blockIdx.x * blockDim.x + threadIdx.x;
    if (i < n) p[i] = v;
}

// ---------------------------------------------------------------------------
// Degree accumulation (self-loop contributes the initial 1.0 in k_fill)
// ---------------------------------------------------------------------------
__global__ void k_deg_accum(const int* __restrict__ dst, float* __restrict__ deg) {
    int e = blockIdx.x * blockDim.x + threadIdx.x;
    if (e < N_EDGES) atomicAdd(&deg[dst[e]], 1.0f);
}

__global__ void k_rsqrt_inplace(float* __restrict__ d, int n) {
    int i = blockIdx.x * blockDim.x + threadIdx.x;
    if (i < n) d[i] = rsqrtf(d[i]);   // deg >= 1 always (self loop), clip is a no-op
}

// ---------------------------------------------------------------------------
// Input embedding: feat[n,d] = weights[n] * lin_in_w[d] + lin_in_b[d]
// ---------------------------------------------------------------------------
__global__ void k_feat(const float* __restrict__ w,
                       const float* __restrict__ lw,
                       const float* __restrict__ lb,
                       float* __restrict__ feat) {
    int i = blockIdx.x * blockDim.x + threadIdx.x;
    if (i < NF) {
        int n = i >> 7, d = i & 127;
        feat[i] = w[n] * lw[d] + lb[d];
    }
}

// ---------------------------------------------------------------------------
// SpMM scatter: agg[dst] += x[src] * dinv[src]   (4 dims per thread, float4 gather)
// Self-loop edges are NOT in the edge list; they are folded into k_finish.
// ---------------------------------------------------------------------------
__global__ void k_scatter(const int* __restrict__ src,
                          const int* __restrict__ dst,
                          const float* __restrict__ x,
                          const float* __restrict__ dinv,
                          float* __restrict__ agg) {
    unsigned t = blockIdx.x * blockDim.x + threadIdx.x;   // E * 32 threads
    if (t >= (unsigned)N_EDGES * 32u) return;
    int e = t >> 5;
    int q = (t & 31) << 2;                                 // dim offset 0..124
    int s = src[e], d = dst[e];
    float sc = dinv[s];
    const float4 v = *(const float4*)(x + (size_t)s * DIM + q);
    float* ap = agg + (size_t)d * DIM + q;
    atomicAdd(ap + 0, v.x * sc);
    atomicAdd(ap + 1, v.y * sc);
    atomicAdd(ap + 2, v.z * sc);
    atomicAdd(ap + 3, v.w * sc);
}

// ---------------------------------------------------------------------------
// Chebyshev recurrence finish:
//   spmm(x)[n,d] = (agg[n,d] + x[n,d]*dinv[n]) * dinv[n]   (self loop folded in)
//   out = alpha * spmm(x) + beta * prev
//   X1: alpha=-1, beta= 0
//   X2: alpha=-2, beta=-1 (prev = X0, x = X1)
// ---------------------------------------------------------------------------
__global__ void k_finish(float* __restrict__ out,
                         const float* __restrict__ agg,
                         const float* __restrict__ x,
                         const float* __restrict__ prev,
                         const float* __restrict__ dinv,
                         float alpha, float beta) {
    int i = blockIdx.x * blockDim.x + threadIdx.x;
    if (i < NF) {
        int n = i >> 7;
        float di = dinv[n];
        float h = (agg[i] + x[i] * di) * di;
        out[i] = alpha * h + beta * prev[i];
    }
}

// ---------------------------------------------------------------------------
// Dense GEMM with WMMA fp32:  out = relu([X0 | X1 | X2] @ W + b)
//   W: [3*DIM, DIM] row-major, b: [DIM]
// One wave (32 lanes) per 16-row node tile; computes all 128 output columns.
// 50000 rows = 3125 * 16 exactly -> no edge masking, EXEC stays all-ones.
//
// v_wmma_f32_16x16x4_f32 operand layout (ISA 7.12.2):
//   A (16x4): lane l(0..15) v0=A[l,k], v1=A[l,k+1]; lane l+16 v0=A[l,k+2], v1=A[l,k+3]
//   B (4x16): mirrored (lane l v0=B[k,l], v1=B[k+1,l]; lane l+16 -> k+2,k+3)
//   C/D (16x16): vgpr r: lanes 0-15 -> D[r, lane], lanes 16-31 -> D[r+8, lane-16]
// ---------------------------------------------------------------------------
__global__ void __launch_bounds__(32)
k_cheb_gemm(const float* __restrict__ X0,
            const float* __restrict__ X1,
            const float* __restrict__ X2,
            const float* __restrict__ W,
            const float* __restrict__ b,
            float* __restrict__ out) {
    const int row0 = blockIdx.x * 16;
    const int lane = threadIdx.x & 31;
    const int half = lane >> 4;     // 0: K={k,k+1}, 1: K={k+2,k+3}
    const int lr   = lane & 15;

    v8f acc[8];
#pragma unroll
    for (int nt = 0; nt < 8; ++nt) {
        float bv = b[nt * 16 + lr];              // bias depends only on column
        acc[nt] = (v8f){bv, bv, bv, bv, bv, bv, bv, bv};
    }

    const float* segs[3] = {X0, X1, X2};
    for (int seg = 0; seg < 3; ++seg) {
        const float* Xp = segs[seg] + (size_t)row0 * DIM;
#pragma unroll 4
        for (int kk = 0; kk < DIM; kk += 4) {
            const int ka = kk + 2 * half;
            v2f a;
            a.x = Xp[(size_t)lr * DIM + ka];
            a.y = Xp[(size_t)lr * DIM + ka + 1];
            const float* Wk  = W + (size_t)(seg * DIM + ka) * DIM;   // row ka
            const float* Wk1 = Wk + DIM;                              // row ka+1
#pragma unroll
            for (int nt = 0; nt < 8; ++nt) {
                v2f bb;
                bb.x = Wk [nt * 16 + lr];
                bb.y = Wk1[nt * 16 + lr];
                acc[nt] = __builtin_amdgcn_wmma_f32_16x16x4_f32(
                    /*neg_a=*/false, a, /*neg_b=*/false, bb,
                    /*c_mod=*/(short)0, acc[nt],
                    /*reuse_a=*/false, /*reuse_b=*/false);
            }
        }
    }

    // ReLU + store (LeakyReLU on a ReLU output is the identity, so it is fused away)
#pragma unroll
    for (int nt = 0; nt < 8; ++nt) {
        const int col = nt * 16 + lr;
#pragma unroll
        for (int r = 0; r < 8; ++r) {
            const int row = row0 + r + 8 * half;
            float v = acc[nt][r];
            out[(size_t)row * DIM + col] = v > 0.0f ? v : 0.0f;
        }
    }
}

// ---------------------------------------------------------------------------
// Prediction head: logits[n] = dot(emb[n,:], pred_w) + pred_b
// ---------------------------------------------------------------------------
__global__ void k_pred(const float* __restrict__ emb,
                       const float* __restrict__ pw,
                       const float* __restrict__ pb,
                       float* __restrict__ out) {
    int n = blockIdx.x * blockDim.x + threadIdx.x;
    if (n < N_NODES) {
        const float* e = emb + (size_t)n * DIM;
        float s = 0.0f;
#pragma unroll 8
        for (int d = 0; d < DIM; d += 4) {
            float4 v = *(const float4*)(e + d);
            s += v.x * pw[d] + v.y * pw[d + 1] + v.z * pw[d + 2] + v.w * pw[d + 3];
        }
        out[n] = s + pb[0];
    }
}

// ---------------------------------------------------------------------------
// Host launcher
// Inputs (setup_inputs order):
//   0 weights [N], 1 src [E], 2 dst [E], 3 lin_in_w [1,128], 4 lin_in_b [128],
//   5 cheb_ws [3,384,128], 6 cheb_bs [3,128], 7 pred_w [128,1], 8 pred_b [1]
// Output: logits [N,1] float32
// ---------------------------------------------------------------------------
extern "C" void kernel_launch(void* const* d_in, const int* in_sizes, int n_in,
                              void* d_out, int out_size, void* d_ws, size_t ws_size,
                              hipStream_t stream) {
    const float* weights  = (const float*)d_in[0];
    const int*   src      = (const int*)  d_in[1];
    const int*   dst      = (const int*)  d_in[2];
    const float* lin_in_w = (const float*)d_in[3];
    const float* lin_in_b = (const float*)d_in[4];
    const float* cheb_ws  = (const float*)d_in[5];
    const float* cheb_bs  = (const float*)d_in[6];
    const float* pred_w   = (const float*)d_in[7];
    const float* pred_b   = (const float*)d_in[8];
    float* logits = (float*)d_out;

    // Workspace layout (floats): dinv | X0 | X1 | X2 | agg | emb  (~128 MB total)
    float* ws   = (float*)d_ws;
    float* dinv = ws;
    float* X0   = ws + 51200;      // 50000 padded up for alignment
    float* X1   = X0 + NF;
    float* X2   = X1 + NF;
    float* agg  = X2 + NF;
    float* emb  = agg + NF;

    const int T = 256;
    const int gN    = (N_NODES + T - 1) / T;
    const int gE    = (N_EDGES + T - 1) / T;
    const int gNF   = (NF + T - 1) / T;
    const int gScat = (N_EDGES * 32 + T - 1) / T;   // 75000 blocks

    // Degrees -> D^{-1/2} (self loop = initial 1.0, so clip(deg,1) is a no-op)
    k_fill<<<gN, T, 0, stream>>>(dinv, 1.0f, N_NODES);
    k_deg_accum<<<gE, T, 0, stream>>>(dst, dinv);
    k_rsqrt_inplace<<<gN, T, 0, stream>>>(dinv, N_NODES);

    // Input embedding
    k_feat<<<gNF, T, 0, stream>>>(weights, lin_in_w, lin_in_b, X0);

    float* cur = X0;   // layer input
    float* nxt = emb;  // layer output (ping-pong)
    for (int layer = 0; layer < 3; ++layer) {
        const float* Wl = cheb_ws + (size_t)layer * 3 * DIM * DIM;
        const float* bl = cheb_bs + (size_t)layer * DIM;

        // X1 = -spmm(cur)
        k_fill<<<gNF, T, 0, stream>>>(agg, 0.0f, NF);
        k_scatter<<<gScat, T, 0, stream>>>(src, dst, cur, dinv, agg);
        k_finish<<<gNF, T, 0, stream>>>(X1, agg, cur, cur, dinv, -1.0f, 0.0f);

        // X2 = -2*spmm(X1) - cur
        k_fill<<<gNF, T, 0, stream>>>(agg, 0.0f, NF);
        k_scatter<<<gScat, T, 0, stream>>>(src, dst, X1, dinv, agg);
        k_finish<<<gNF, T, 0, stream>>>(X2, agg, X1, cur, dinv, -2.0f, -1.0f);

        // nxt = relu([cur|X1|X2] @ Wl + bl)   -- WMMA fp32 path
        k_cheb_gemm<<<N_NODES / 16, 32, 0, stream>>>(cur, X1, X2, Wl, bl, nxt);

        float* t = cur; cur = nxt; nxt = t;
    }

    k_pred<<<gN, T, 0, stream>>>(cur, pred_w, pred_b, logits);
}